// PizzaBurningEffect_19516331393067
// MI455X (gfx1250) — compile-verified
//
#include <hip/hip_runtime.h>

typedef float v4f __attribute__((ext_vector_type(4)));
typedef int   v4i __attribute__((ext_vector_type(4)));

#define AS1 __attribute__((address_space(1)))
#define AS3 __attribute__((address_space(3)))

// B=32, C=3, H=512, W=512, S=8
#define HW  262144      // 512*512
#define STEPF (2.0f / 511.0f)

// Analytic ew normalization constants (grid extremes of dist = sqrt(xc^2+yc^2)
// on linspace(-1,1,512)^2: min = sqrt(2)/511, max = sqrt(2)):
//   ewmin = exp(2*sqrt(2)/511 - 1.4) , ewmax = exp(2*sqrt(2) - 1.4)
//   invrange = 1/(ewmax - ewmin + 1e-6)
#define EWMIN    0.24796578f
#define INVRANGE 0.25475215f   // 1/(4.1723485 - 0.24796578 + 1e-6)

// ---------------------------------------------------------------------------
// Hot kernel: one thread = 4 consecutive pixels (same row) x 3 channels.
// grid = (256 tiles, 32 batches), block = 256 threads (8 wave32s).
// Spot params for this batch are async-DMA'd into LDS (CDNA5 path), image is
// loaded RT (stays resident in 192MB L2 across replays), output stored NT.
// Roofline: 201 MB traffic -> ~8.6us at 23.3 TB/s; per-pixel math reduced to
// 2 exps + 1 sqrt via log-domain spot max.
// ---------------------------------------------------------------------------
__global__ __launch_bounds__(256) void burn_kernel(
    const float* __restrict__ img, const v4f* __restrict__ spw,
    const float* __restrict__ burnw, float* __restrict__ out)
{
    __shared__ v4f sp[8];
    const int b   = blockIdx.y;
    const int tid = threadIdx.x;

    // --- async global->LDS copy of this batch's 8 spot-param float4s -------
    if (tid < 32) {                       // wave 0 only: EXEC all-ones in wave
        const int s = tid & 7;            // lanes 8..31 duplicate (benign)
#if __has_builtin(__builtin_amdgcn_global_load_async_to_lds_b128)
        __builtin_amdgcn_global_load_async_to_lds_b128(
            (AS1 v4i*)(spw + (b << 3) + s),
            (AS3 v4i*)&sp[s], 0, 0);
#else
        {
            unsigned lds_off =
                (unsigned)(unsigned long long)(AS3 v4f*)&sp[s];
            unsigned long long gaddr =
                (unsigned long long)(const void*)(spw + (b << 3) + s);
            asm volatile("global_load_async_to_lds_b128 %0, %1, off"
                         :: "v"(lds_off), "v"(gaddr) : "memory");
        }
#endif
#if __has_builtin(__builtin_amdgcn_s_wait_asynccnt)
        __builtin_amdgcn_s_wait_asynccnt(0);
#else
        asm volatile("s_wait_asynccnt 0" ::: "memory");
#endif
    }
    __syncthreads();

    const float burn = burnw[b];

    const int p0 = (blockIdx.x * 256 + tid) * 4;   // flat pixel index in HxW
    const int h  = p0 >> 9;
    const int w0 = p0 & 511;
    const float yc  = fmaf((float)h, STEPF, -1.0f);
    const float yc2 = yc * yc;

    // Hoist per-spot terms (yc fixed for all 4 pixels of this thread)
    float sxA[8], dy2A[8], nzA[8], lwA[8];
#pragma unroll
    for (int s = 0; s < 8; ++s) {
        const v4f P = sp[s];
        sxA[s] = P.x;
        const float dy = yc - P.y;
        dy2A[s] = dy * dy;
        nzA[s]  = -P.z;           // -(1/(2r^2))
        lwA[s]  = P.w;            // log(sint)
    }

    float bm[4];
#pragma unroll
    for (int j = 0; j < 4; ++j) {
        const float xc   = fmaf((float)(w0 + j), STEPF, -1.0f);
        const float dist = __builtin_amdgcn_sqrtf(fmaf(xc, xc, yc2));
        const float ewn  = (__expf(fmaf(2.0f, dist, -1.4f)) - EWMIN) * INVRANGE;
        float m = -3.0e38f;
#pragma unroll
        for (int s = 0; s < 8; ++s) {
            const float dx = xc - sxA[s];
            const float d2 = fmaf(dx, dx, dy2A[s]);
            m = fmaxf(m, fmaf(d2, nzA[s], lwA[s]));    // -d2/(2r^2)+log(i)
        }
        const float spotv = __expf(m);
        bm[j] = fminf(fmaxf(fmaxf(ewn, spotv) * burn, 0.0f), 1.0f);
    }

    // out = clip(img * (1 - bm*(1-dark_scale_c)), 0, 1)
    const size_t base = (size_t)b * (3u * HW) + (size_t)p0;
#pragma unroll
    for (int c = 0; c < 3; ++c) {
        const float k = (c == 0) ? 0.3f : (c == 1) ? 0.6f : 0.7f; // 1-ds
        const v4f px = *(const v4f*)(img + base + (size_t)c * HW);
        v4f o;
        o.x = fminf(fmaxf(px.x * fmaf(-bm[0], k, 1.0f), 0.0f), 1.0f);
        o.y = fminf(fmaxf(px.y * fmaf(-bm[1], k, 1.0f), 0.0f), 1.0f);
        o.z = fminf(fmaxf(px.z * fmaf(-bm[2], k, 1.0f), 0.0f), 1.0f);
        o.w = fminf(fmaxf(px.w * fmaf(-bm[3], k, 1.0f), 0.0f), 1.0f);
        __builtin_nontemporal_store(o, (v4f*)(out + base + (size_t)c * HW));
    }
}

// ---------------------------------------------------------------------------
// Kernel A: precompute per-(b,s) spot parameters into d_ws.
//   spw[b*8+s] = { sx, sy, 1/(2r^2), log(sint) }   (float4, 4 KB total)
//   burnw[b]   = 0.2 + 0.6*u_burn[b]               (128 B)
// Using log(sint) lets the main kernel take max in log-domain -> ONE exp for
// all 8 spots instead of 8 exps per pixel. 256 threads total: negligible.
// ---------------------------------------------------------------------------
__global__ __launch_bounds__(256) void spot_params_kernel(
    const float* __restrict__ u_xy, const float* __restrict__ u_radius,
    const float* __restrict__ u_intensity, const float* __restrict__ u_burn,
    v4f* __restrict__ spw, float* __restrict__ burnw)
{
    const int i = threadIdx.x;                        // i = b*8 + s, 256 threads
    const float sx = fmaf(2.0f, u_xy[2 * i + 0], -1.0f);
    const float sy = fmaf(2.0f, u_xy[2 * i + 1], -1.0f);
    const float r  = fmaf(0.15f, u_radius[i], 0.05f);
    const float inv2r2 = 1.0f / (2.0f * r * r);
    const float li = __logf(fmaf(0.5f, u_intensity[i], 0.5f));
    v4f P; P.x = sx; P.y = sy; P.z = inv2r2; P.w = li;
    spw[i] = P;
    if ((i & 7) == 0) {
        const int b = i >> 3;
        burnw[b] = fmaf(0.6f, u_burn[b], 0.2f);
    }
}

// ---------------------------------------------------------------------------
extern "C" void kernel_launch(void* const* d_in, const int* in_sizes, int n_in,
                              void* d_out, int out_size, void* d_ws, size_t ws_size,
                              hipStream_t stream)
{
    const float* img    = (const float*)d_in[0];   // (32,3,512,512)
    const float* u_xy   = (const float*)d_in[1];   // (32,8,2)
    const float* u_rad  = (const float*)d_in[2];   // (32,8)
    const float* u_int  = (const float*)d_in[3];   // (32,8)
    const float* u_burn = (const float*)d_in[4];   // (32,)
    float* out = (float*)d_out;

    v4f*   spw   = (v4f*)d_ws;                          // 256 * 16B = 4 KB
    float* burnw = (float*)((char*)d_ws + 256 * sizeof(v4f)); // + 128 B

    spot_params_kernel<<<1, 256, 0, stream>>>(u_xy, u_rad, u_int, u_burn,
                                              spw, burnw);

    burn_kernel<<<dim3(256, 32), 256, 0, stream>>>(img, (const v4f*)spw,
                                                   (const float*)burnw, out);

    (void)in_sizes; (void)n_in; (void)out_size; (void)ws_size;
}